// VectorQuantizerEMA_1365799600184
// MI455X (gfx1250) — compile-verified
//
#include <hip/hip_runtime.h>
#include <hip/hip_bf16.h>
#include <math.h>

typedef __attribute__((ext_vector_type(2))) float v2f;
typedef __attribute__((ext_vector_type(8))) float v8f;

// ---- problem geometry ----
#define N_E    1024
#define E_DIM  64
#define N_VEC  65536     // 16*64*64
// ---- output layout (floats) ----
#define OUT_LOSS   0
#define OUT_ZQ     1                      // 4194304 elements
#define OUT_PERP   4194305
#define OUT_ENC    4194306                // 67108864 elements
#define OUT_IDX    71303170               // 65536 elements

// ========================= init: histogram zero + ||e||^2 =========================
__global__ __launch_bounds__(256) void vq_init(const float* __restrict__ emb,
                                               unsigned int* __restrict__ counts,
                                               float* __restrict__ e_norm) {
    int j = blockIdx.x * 256 + threadIdx.x;
    if (j < N_E) {
        counts[j] = 0u;
        float s = 0.f;
        #pragma unroll
        for (int c = 0; c < E_DIM; ++c) { float e = emb[j * E_DIM + c]; s += e * e; }
        e_norm[j] = s;
    }
}

// ========================= main: argmin via f32 WMMA + outputs =========================
__global__ __launch_bounds__(256) void vq_main(const float* __restrict__ z,
                                               const float* __restrict__ emb,
                                               const float* __restrict__ e_norm,
                                               unsigned int* __restrict__ counts,
                                               float* __restrict__ partial_loss,
                                               float* __restrict__ out) {
    __shared__ float lds_z[64][65];               // [w][c], padded vs bank conflicts
    __shared__ unsigned long long lds_min[64];    // packed (sortkey<<32 | index)
    __shared__ int lds_idx[64];
    __shared__ float red[256];

    const int wg  = blockIdx.x;          // 0..1023 : one (b,h) slice
    const int b   = wg >> 6;
    const int h   = wg & 63;
    const int tid = threadIdx.x;

    // ---- load z[b, :, h, :] -> lds_z[w][c] (transpose to row-major A) ----
    #pragma unroll
    for (int it = 0; it < 16; ++it) {
        int flat = it * 256 + tid;                 // 4096 elements
        int c = flat >> 6, w = flat & 63;
        lds_z[w][c] = z[((size_t)(b * 64 + c) * 64 + h) * 64 + w];
    }
    if (tid < 64) lds_min[tid] = ~0ull;
    __syncthreads();

    const int lane    = tid & 31;
    const int wave    = tid >> 5;
    const int strip   = wave & 3;        // 16-row strip: rows strip*16 .. +15
    const int colhalf = wave >> 2;       // 512-code half

    // ---- A fragments: 16x4 f32 layout: lanes0-15 M=lane, vgpr{0,1}=K{0,1}; lanes16-31 K{2,3}
    const int m  = strip * 16 + (lane & 15);
    const int kb = 2 * (lane >> 4);
    v2f afrag[16];
    #pragma unroll
    for (int k = 0; k < 16; ++k) {
        afrag[k].x = lds_z[m][k * 4 + kb];
        afrag[k].y = lds_z[m][k * 4 + kb + 1];
    }

    float minv[8];
    int   mini[8];
    #pragma unroll
    for (int i = 0; i < 8; ++i) { minv[i] = 3.4e38f; mini[i] = 0; }

    const int ncol = lane & 15;          // this lane's column within each 16-code tile
    for (int t = 0; t < 32; ++t) {
        const int code0 = (colhalf * 32 + t) * 16;
        const int code  = code0 + ncol;
        // B fragments: 4x16 f32: lanes0-15 N=lane K{0,1}; lanes16-31 N=lane-16 K{2,3}
        const float* brow = emb + (size_t)code * E_DIM + kb;
        if (t < 31) __builtin_prefetch(brow + E_DIM * 16, 0, 3);  // next tile's row

        // batch all 16 B-fragment loads so they clause with one wait
        v2f bfrag[16];
        #pragma unroll
        for (int k = 0; k < 16; ++k) {
            bfrag[k].x = brow[k * 4];
            bfrag[k].y = brow[k * 4 + 1];
        }
        const float en = e_norm[code];

        // two accumulator chains to keep the WMMA pipe fed
        v8f acc0 = {};
        v8f acc1 = {};
        #pragma unroll
        for (int k = 0; k < 8; ++k) {
            acc0 = __builtin_amdgcn_wmma_f32_16x16x4_f32(
                false, afrag[2 * k], false, bfrag[2 * k], (short)0, acc0, false, false);
            acc1 = __builtin_amdgcn_wmma_f32_16x16x4_f32(
                false, afrag[2 * k + 1], false, bfrag[2 * k + 1], (short)0, acc1, false, false);
        }

        #pragma unroll
        for (int i = 0; i < 8; ++i) {
            float s = acc0[i] + acc1[i];
            float d = en - 2.0f * s;               // ||e||^2 - 2 z.e  (drop row-const ||z||^2)
            if (d < minv[i]) { minv[i] = d; mini[i] = code; }
        }
    }

    // ---- cross-lane / cross-wave argmin via packed u64 LDS atomicMin ----
    const int rbase = strip * 16 + 8 * (lane >> 4);
    #pragma unroll
    for (int i = 0; i < 8; ++i) {
        unsigned int fb  = __float_as_uint(minv[i]);
        unsigned int key = fb ^ ((unsigned int)((int)fb >> 31) | 0x80000000u); // monotonic map
        unsigned long long packed = ((unsigned long long)key << 32) | (unsigned int)mini[i];
        atomicMin(&lds_min[rbase + i], packed);
    }
    __syncthreads();
    if (tid < 64) lds_idx[tid] = (int)(lds_min[tid] & 0xFFFFFFFFu);
    __syncthreads();

    // ---- indices output + histogram ----
    if (tid < 64) {
        int n = b * 4096 + h * 64 + tid;            // n = b*H*W + h*W + w
        out[OUT_IDX + n] = (float)lds_idx[tid];
        atomicAdd(&counts[lds_idx[tid]], 1u);
    }

    // ---- z_q_st (== z_q values) in (B,C,H,W) layout + commitment-loss partial ----
    float lsum = 0.f;
    #pragma unroll
    for (int it = 0; it < 16; ++it) {
        int flat = it * 256 + tid;
        int c = flat >> 6, w = flat & 63;
        float zq = emb[(size_t)lds_idx[w] * E_DIM + c];
        float zv = lds_z[w][c];
        float df = zq - zv;
        lsum += df * df;
        __builtin_nontemporal_store(zq, &out[OUT_ZQ + ((size_t)(b * 64 + c) * 64 + h) * 64 + w]);
    }
    red[tid] = lsum;
    __syncthreads();
    #pragma unroll
    for (int s = 128; s > 0; s >>= 1) {
        if (tid < s) red[tid] += red[tid + s];
        __syncthreads();
    }
    if (tid == 0) partial_loss[wg] = red[0];

    // ---- one-hot min_encodings: 64 rows x 1024, nontemporal 8B vector stores ----
    for (int it = 0; it < 128; ++it) {
        int flat = it * 256 + tid;                 // 32768 float2 stores
        int w  = flat >> 9;                        // 512 float2 per row
        int j2 = flat & 511;
        int n  = b * 4096 + h * 64 + w;
        int idx = lds_idx[w];
        v2f v;
        v.x = (j2 * 2     == idx) ? 1.0f : 0.0f;
        v.y = (j2 * 2 + 1 == idx) ? 1.0f : 0.0f;
        v2f* p = (v2f*)&out[OUT_ENC + (size_t)n * 1024 + j2 * 2]; // 8B aligned
        __builtin_nontemporal_store(v, p);
    }
}

// ========================= final: loss + perplexity scalars =========================
__global__ __launch_bounds__(256) void vq_final(const unsigned int* __restrict__ counts,
                                                const float* __restrict__ partial_loss,
                                                float* __restrict__ out) {
    __shared__ float redl[256];
    __shared__ float rede[256];
    int tid = threadIdx.x;
    float ls = 0.f, ent = 0.f;
    for (int i = tid; i < 1024; i += 256) ls += partial_loss[i];
    for (int j = tid; j < N_E; j += 256) {
        float p = (float)counts[j] * (1.0f / 65536.0f);
        ent += p * logf(p + 1e-10f);
    }
    redl[tid] = ls; rede[tid] = ent;
    __syncthreads();
    #pragma unroll
    for (int s = 128; s > 0; s >>= 1) {
        if (tid < s) { redl[tid] += redl[tid + s]; rede[tid] += rede[tid + s]; }
        __syncthreads();
    }
    if (tid == 0) {
        out[OUT_LOSS] = 0.25f * redl[0] / 4194304.0f;  // BETA * mean((zq - z)^2)
        out[OUT_PERP] = expf(-rede[0]);
    }
}

extern "C" void kernel_launch(void* const* d_in, const int* in_sizes, int n_in,
                              void* d_out, int out_size, void* d_ws, size_t ws_size,
                              hipStream_t stream) {
    const float* z   = (const float*)d_in[0];   // (16,64,64,64) f32
    const float* emb = (const float*)d_in[1];   // (1024,64) f32
    float* out = (float*)d_out;

    unsigned int* counts  = (unsigned int*)d_ws;          // 1024 u32
    float*        e_norm  = (float*)d_ws + 1024;          // 1024 f32
    float*        partial = (float*)d_ws + 2048;          // 1024 f32

    vq_init<<<4, 256, 0, stream>>>(emb, counts, e_norm);
    vq_main<<<1024, 256, 0, stream>>>(z, emb, e_norm, counts, partial, out);
    vq_final<<<1, 256, 0, stream>>>(counts, partial, out);
}